// GCN_novel_84327387889926
// MI455X (gfx1250) — compile-verified
//
#include <hip/hip_runtime.h>
#include <hip/hip_bf16.h>

// CDNA5 wave32 WMMA fragment types
typedef __attribute__((ext_vector_type(2))) float v2f;
typedef __attribute__((ext_vector_type(8))) float v8f;

// -------------------- degree / normalization --------------------

__global__ void fill_ones_f32(float* __restrict__ p, int n) {
    int i = blockIdx.x * blockDim.x + threadIdx.x;
    if (i < n) p[i] = 1.0f;   // self-loop contributes 1 to every node's degree
}

__global__ void deg_accum(float* __restrict__ deg, const int* __restrict__ dst, int e_cnt) {
    int i = blockIdx.x * blockDim.x + threadIdx.x;
    if (i < e_cnt) unsafeAtomicAdd(&deg[dst[i]], 1.0f);
}

__global__ void rsqrt_inplace(float* __restrict__ p, int n) {
    int i = blockIdx.x * blockDim.x + threadIdx.x;
    if (i < n) p[i] = rsqrtf(p[i]);   // deg >= 1 always (self-loops), no zero case
}

__global__ void norm_compute(float* __restrict__ nrm, const float* __restrict__ dis,
                             const int* __restrict__ src, const int* __restrict__ dst, int e_cnt) {
    int i = blockIdx.x * blockDim.x + threadIdx.x;
    if (i < e_cnt) nrm[i] = dis[src[i]] * dis[dst[i]];
}

// -------------------- weight padding (ragged d_out only; layer 5: 40 -> 48) ------------

__global__ void pad_weights(const float* __restrict__ W, float* __restrict__ Wp,
                            int K, int D, int Dpad) {
    int i = blockIdx.x * blockDim.x + threadIdx.x;
    if (i >= K * Dpad) return;
    int k = i / Dpad;
    int c = i - k * Dpad;
    Wp[i] = (c < D) ? W[k * D + c] : 0.0f;
}

// -------------------- dense GEMM via V_WMMA_F32_16X16X4_F32 --------------------
// One wave computes one 16x16 tile of H = X @ W (fp32, full precision).
// A 16x4 fp32 fragment: lanes 0-15 hold M=lane,   K = k+0 / k+1;
//                       lanes 16-31 hold M=lane-16, K = k+2 / k+3.
// B 4x16 fragment is the symmetric layout over N. W is pre-padded so the
// K-loop is completely branch-free (no exec-mask churn around the B loads).
// C/D: 8 VGPRs, lane<16 -> rows M=0..7 (v0..v7), lane>=16 -> rows M=8..15.

template <bool RELU>
__global__ __launch_bounds__(128) void gcn_gemm_wmma(
        const float* __restrict__ X, const float* __restrict__ W,
        float* __restrict__ H, int K, int Nout, int ldw /* padded d_out, %16==0 */)
{
    const int lane = threadIdx.x;          // 0..31, wave32
    const int half = lane >> 4;            // 0 or 1
    const int l15  = lane & 15;
    const int ntiles = ldw >> 4;
    const int tn = blockIdx.y * 4 + threadIdx.y;   // N-tile, wave-uniform
    if (tn >= ntiles) return;                      // whole-wave exit; EXEC all-1s for WMMA
    const int tm  = blockIdx.x;                    // M-tile (M is an exact multiple of 16)
    const int row = tm * 16 + l15;                 // A-operand row for this lane
    const int col = tn * 16 + l15;                 // B/D column for this lane

    const float* __restrict__ xrow = X + (size_t)row * K + half * 2;       // a0,a1 contiguous
    const float* __restrict__ wcol = W + (size_t)(half * 2) * ldw + col;   // b0,b1 stride ldw

    v8f acc = {};
#pragma unroll 4
    for (int k = 0; k < K; k += 4) {
        v2f av, bv;
        av.x = xrow[0];
        av.y = xrow[1];
        if (RELU) { av.x = fmaxf(av.x, 0.0f); av.y = fmaxf(av.y, 0.0f); }  // fused prev-layer ReLU
        bv.x = wcol[0];
        bv.y = wcol[ldw];
        // (neg_a, A, neg_b, B, c_mod, C, reuse_a, reuse_b)
        acc = __builtin_amdgcn_wmma_f32_16x16x4_f32(
                false, av, false, bv, (short)0, acc, false, false);
        xrow += 4;
        wcol += (size_t)4 * ldw;
    }

    if (col < Nout) {                               // only the last tile of d_out=40 is partial
        const int rbase = tm * 16 + half * 8;
#pragma unroll
        for (int v = 0; v < 8; ++v)
            H[(size_t)(rbase + v) * Nout + col] = acc[v];
    }
}

// -------------------- self-loop + bias init (full overwrite of Y) --------------------
// Y[i,c] = H[i,c] * dis[i]^2 + b[c]   (dis^2 == 1/deg, the self-loop's norm)

__global__ void init_selfloop_bias(const float* __restrict__ H, const float* __restrict__ dis,
                                   const float* __restrict__ bias, float* __restrict__ Y,
                                   int n, int dout)
{
    const int vec = dout >> 2;                      // dout is always a multiple of 4
    long long tid = blockIdx.x * (long long)blockDim.x + threadIdx.x;
    if (tid >= (long long)n * vec) return;
    const int i  = (int)(tid / vec);
    const int c4 = (int)(tid - (long long)i * vec) * 4;
    const float d = dis[i];
    const float s = d * d;
    const float4 hv = *(const float4*)(H + (size_t)i * dout + c4);
    const float4 bv = *(const float4*)(bias + c4);
    float4 o;
    o.x = hv.x * s + bv.x;
    o.y = hv.y * s + bv.y;
    o.z = hv.z * s + bv.z;
    o.w = hv.w * s + bv.w;
    *(float4*)(Y + (size_t)i * dout + c4) = o;
}

// -------------------- edge scatter-add (L2-resident gather + global_atomic_add_f32) ----

__global__ void agg_edges(const float* __restrict__ H, const float* __restrict__ nrm,
                          const int* __restrict__ src, const int* __restrict__ dst,
                          float* __restrict__ Y, int e_cnt, int dout)
{
    const int vec = dout >> 2;
    long long tid = blockIdx.x * (long long)blockDim.x + threadIdx.x;
    if (tid >= (long long)e_cnt * vec) return;
    const int e  = (int)(tid / vec);
    const int c4 = (int)(tid - (long long)e * vec) * 4;
    const int s = src[e];
    const int d = dst[e];
    const float w = nrm[e];
    const float4 hv = *(const float4*)(H + (size_t)s * dout + c4);
    float* yp = Y + (size_t)d * dout + c4;
    unsafeAtomicAdd(yp + 0, hv.x * w);
    unsafeAtomicAdd(yp + 1, hv.y * w);
    unsafeAtomicAdd(yp + 2, hv.z * w);
    unsafeAtomicAdd(yp + 3, hv.w * w);
}

// -------------------- launch --------------------

extern "C" void kernel_launch(void* const* d_in, const int* in_sizes, int n_in,
                              void* d_out, int out_size, void* d_ws, size_t ws_size,
                              hipStream_t stream)
{
    const int N = in_sizes[0] / 512;   // 50000
    const int E = in_sizes[1] / 2;     // 800000

    const float* x  = (const float*)d_in[0];
    const int*   ei = (const int*)d_in[1];
    const int*   src = ei;             // edge_index[0]
    const int*   dst = ei + E;         // edge_index[1]
    const float* Wl[5] = { (const float*)d_in[2], (const float*)d_in[4], (const float*)d_in[6],
                           (const float*)d_in[8], (const float*)d_in[10] };
    const float* bl[5] = { (const float*)d_in[3], (const float*)d_in[5], (const float*)d_in[7],
                           (const float*)d_in[9], (const float*)d_in[11] };
    const int dims[6] = {512, 64, 256, 32, 16, 40};

    // workspace layout (floats): dis | norm | bufA (h, N*256) | bufB (N*256) | bufC (N*64) | Wpad
    float* degdis = (float*)d_ws;                 // N
    float* nrm    = degdis + N;                   // E
    float* bufA   = nrm + E;                      // N*256  (GEMM output h, every layer)
    float* bufB   = bufA + (size_t)N * 256;       // N*256  (y2)
    float* bufC   = bufB + (size_t)N * 256;       // N*64   (y1/y3/y4)
    float* wpad   = bufC + (size_t)N * 64;        // 512*256 max (padded weights)

    // symmetric GCN normalization, computed once and reused by all 5 layers
    fill_ones_f32 <<<(N + 255) / 256, 256, 0, stream>>>(degdis, N);
    deg_accum     <<<(E + 255) / 256, 256, 0, stream>>>(degdis, dst, E);
    rsqrt_inplace <<<(N + 255) / 256, 256, 0, stream>>>(degdis, N);
    norm_compute  <<<(E + 255) / 256, 256, 0, stream>>>(nrm, degdis, src, dst, E);

    // buffer rotation: h always in bufA; y alternates so the GEMM input is never overwritten
    const float* in_ptr[5] = { x, bufC, bufB, bufC, bufC };
    float*       y_ptr[5]  = { bufC, bufB, bufC, bufC, (float*)d_out };

    for (int l = 0; l < 5; ++l) {
        const int K = dims[l], D = dims[l + 1];
        const int Dpad = (D + 15) & ~15;

        const float* Wg = Wl[l];
        if (Dpad != D) {   // only layer 5 (40 -> 48): zero-pad so the GEMM loop is branch-free
            pad_weights<<<(K * Dpad + 255) / 256, 256, 0, stream>>>(Wl[l], wpad, K, D, Dpad);
            Wg = wpad;
        }

        const int ntiles = Dpad / 16;
        dim3 ggrid((N + 15) / 16, (ntiles + 3) / 4);
        dim3 gblk(32, 4);   // 4 waves per block, one 16x16 tile each
        if (l == 0)
            gcn_gemm_wmma<false><<<ggrid, gblk, 0, stream>>>(in_ptr[l], Wg, bufA, K, D, Dpad);
        else
            gcn_gemm_wmma<true ><<<ggrid, gblk, 0, stream>>>(in_ptr[l], Wg, bufA, K, D, Dpad);

        const int vec = D / 4;
        long long ntot = (long long)N * vec;
        init_selfloop_bias<<<(unsigned)((ntot + 255) / 256), 256, 0, stream>>>(
                bufA, degdis, bl[l], y_ptr[l], N, D);

        long long etot = (long long)E * vec;
        agg_edges<<<(unsigned)((etot + 255) / 256), 256, 0, stream>>>(
                bufA, nrm, src, dst, y_ptr[l], E, D);
    }
}